// Lut3D_27161373180057
// MI455X (gfx1250) — compile-verified
//
#include <hip/hip_runtime.h>
#include <cstdint>

#define LUT_DIM   33
#define LUT_N     35937            // 33^3
#define IMG_HW    (1080 * 1920)
#define PIXELS    (8 * IMG_HW)
#define QUADS     (PIXELS / 4)     // 4 pixels per thread-iteration
#define HW_QUADS  (IMG_HW / 4)
// packed RG table: (LUT_N + 1 pad) float2 entries = 287,504 bytes
#define RG_BYTES  ((size_t)(LUT_N + 1) * sizeof(float2))
#define RG_ELEMS  (LUT_N + 1)              // 35938 float2 (8B) elements
#define ELEMS_PER_WAVE 1124                // 32 waves * 1124 >= 35938

typedef unsigned int v4u __attribute__((ext_vector_type(4)));
typedef int          v4i __attribute__((ext_vector_type(4)));
typedef int          v8i __attribute__((ext_vector_type(8)));
typedef float        v4f __attribute__((ext_vector_type(4)));

typedef __attribute__((address_space(3))) char lds_char;

#if __has_builtin(__builtin_amdgcn_tensor_load_to_lds) && \
    __has_builtin(__builtin_amdgcn_s_wait_tensorcnt)
#define USE_TDM 1
#pragma message("CDNA5 probe: tensor_load_to_lds path ENABLED")
#else
#define USE_TDM 0
#pragma message("CDNA5 probe: tensor_load_to_lds path DISABLED (builtin missing)")
#endif

// ---------------------------------------------------------------------------
// Prepack: interleave R and G LUT channels into float2 so corner gathers are
// single aligned ds_load_b64 pairs from LDS. Blue stays in the original LUT
// (contiguous in ir, gathered through WGP$/L2).
// ---------------------------------------------------------------------------
__global__ void lut_prepack_kernel(const float* __restrict__ lut,
                                   float2* __restrict__ rg) {
    int i = blockIdx.x * blockDim.x + threadIdx.x;
    if (i < LUT_N) {
        rg[i] = make_float2(lut[i], lut[LUT_N + i]);
    }
    if (i == 0) rg[LUT_N] = make_float2(0.0f, 0.0f);   // pad entry
}

// ---------------------------------------------------------------------------
// Per-pixel trilinear interpolation against LDS RG table + cached blue LUT.
// ---------------------------------------------------------------------------
__device__ __forceinline__ void interp_one(float vr, float vg, float vb,
                                           const float2* __restrict__ s_rg,
                                           const float* __restrict__ lutB,
                                           float& outr, float& outg, float& outb) {
    const float binsize = (float)(1.000001 / 32.0);   // matches reference f64->f32

    float xr  = vr / binsize;  int ir = (int)floorf(xr);  float fr = xr  - (float)ir;
    float xg  = vg / binsize;  int ig = (int)floorf(xg);  float fg = xg  - (float)ig;
    float xbv = vb / binsize;  int ib = (int)floorf(xbv); float fb = xbv - (float)ib;

    const float wr0 = 1.0f - fr, wr1 = fr;
    const float wg0 = 1.0f - fg, wg1 = fg;
    const float wb0 = 1.0f - fb, wb1 = fb;

    float accr = 0.0f, accg = 0.0f, accb = 0.0f;
    const int base = (ib * LUT_DIM + ig) * LUT_DIM + ir;

#pragma unroll
    for (int k = 0; k < 4; ++k) {
        const int db = k >> 1, dg = k & 1;
        const int e = base + db * (LUT_DIM * LUT_DIM) + dg * LUT_DIM;

        float2 c0 = s_rg[e];       // ds_load_b64 (8B aligned)
        float2 c1 = s_rg[e + 1];   // ds_load_b64
        float  b0 = lutB[e];       // cached WGP$/L2 gather
        float  b1 = lutB[e + 1];

        const float w  = (db ? wb1 : wb0) * (dg ? wg1 : wg0);
        const float w0 = w * wr0;
        const float w1 = w * wr1;
        accr += w0 * c0.x + w1 * c1.x;
        accg += w0 * c0.y + w1 * c1.y;
        accb += w0 * b0   + w1 * b1;
    }
    outr = accr; outg = accg; outb = accb;
}

// ---------------------------------------------------------------------------
// Main kernel: stage RG LUT into LDS via the Tensor Data Mover (one D# per
// wave covering a 1-row tile of its slice), then grid-stride trilinear interp
// over 4-pixel quads (b128 NT streaming loads/stores).
// ---------------------------------------------------------------------------
__global__ void __launch_bounds__(1024)
lut3d_apply_kernel(const float* __restrict__ lut,
                   const float* __restrict__ x,
                   float* __restrict__ out,
                   const float2* __restrict__ ws_rg,
                   int use_ws) {
    extern __shared__ float2 s_rg[];   // RG_ELEMS float2 = 287,504 B
    const int tid  = threadIdx.x;
    const int nthr = blockDim.x;

    if (use_ws) {
#if USE_TDM
        // ---- TDM staging: one tensor_load_to_lds per wave -----------------
        const int wave = tid >> 5;                  // wave32
        int start = wave * ELEMS_PER_WAVE;
        int count = RG_ELEMS - start;
        if (count > ELEMS_PER_WAVE) count = ELEMS_PER_WAVE;
        if (count > 0) {
            const uint64_t gaddr =
                (uint64_t)(uintptr_t)ws_rg + (uint64_t)start * 8u;
            const uint32_t laddr =
                (uint32_t)(uintptr_t)(lds_char*)(void*)s_rg + (uint32_t)start * 8u;

            // D# group 0: count=1 valid, lds_addr, global_addr[56:0], type=2
            v4u g0;
            g0.x = 0x1u;                                   // count=1, flags=0
            g0.y = laddr;                                  // lds_addr (bytes)
            g0.z = (uint32_t)(gaddr & 0xFFFFFFFFu);        // global_addr[31:0]
            g0.w = (uint32_t)((gaddr >> 32) & 0x01FFFFFFu) // global_addr[56:32]
                 | 0x80000000u;                            // type=2 at [127:126]

            // D# group 1: data_size=3 (8B), 1 row of `count` elements
            v8i g1;
            g1[0] = (int)(3u << 16);                          // wg_mask=0, data_size=3
            g1[1] = (int)(((uint32_t)count & 0xFFFFu) << 16); // tensor_dim0 lo16
            g1[2] = (int)((((uint32_t)count >> 16) & 0xFFFFu)
                 | (1u << 16));                               // td0 hi16 | tensor_dim1=1
            g1[3] = (int)(((uint32_t)count & 0xFFFFu) << 16); // tile_dim0
            g1[4] = 1;                                        // tile_dim1=1, tile_dim2=0
            g1[5] = count;                                    // tensor_dim0_stride lo32
            g1[6] = 0;
            g1[7] = 0;

            v4i gz = {0, 0, 0, 0};                            // groups 2/3 unused (2D)
#if defined(__clang_major__) && (__clang_major__ >= 23)
            v8i gz8 = {0, 0, 0, 0, 0, 0, 0, 0};
            __builtin_amdgcn_tensor_load_to_lds(g0, g1, gz, gz, gz8, 0);
#else
            __builtin_amdgcn_tensor_load_to_lds(g0, g1, gz, gz, 0);
#endif
        }
        __builtin_amdgcn_s_wait_tensorcnt(0);
#else
        for (int i = tid; i < RG_ELEMS; i += nthr) s_rg[i] = ws_rg[i];
#endif
    } else {
        // Fallback staging: interleave directly from the original LUT.
        for (int i = tid; i < LUT_N; i += nthr)
            s_rg[i] = make_float2(lut[i], lut[LUT_N + i]);
        if (tid == 0) s_rg[LUT_N] = make_float2(0.0f, 0.0f);
    }
    __syncthreads();

    const float* __restrict__ lutB = lut + 2 * LUT_N;

    const int stride = gridDim.x * nthr;
    for (int q = blockIdx.x * nthr + tid; q < QUADS; q += stride) {
        const int b   = q / HW_QUADS;
        const int hwq = q - b * HW_QUADS;
        const float* __restrict__ xplane = x + (size_t)b * 3 * IMG_HW;

        // 16B-aligned non-temporal b128 streaming loads (one per plane).
        v4f vr4 = __builtin_nontemporal_load((const v4f*)xplane + hwq);
        v4f vg4 = __builtin_nontemporal_load((const v4f*)(xplane + IMG_HW) + hwq);
        v4f vb4 = __builtin_nontemporal_load((const v4f*)(xplane + 2 * IMG_HW) + hwq);

        v4f or4, og4, ob4;
#pragma unroll
        for (int j = 0; j < 4; ++j) {
            float r, g, bch;
            interp_one(vr4[j], vg4[j], vb4[j], s_rg, lutB, r, g, bch);
            or4[j] = r; og4[j] = g; ob4[j] = bch;
        }

        float* __restrict__ oplane = out + (size_t)b * 3 * IMG_HW;
        __builtin_nontemporal_store(or4, (v4f*)oplane + hwq);
        __builtin_nontemporal_store(og4, (v4f*)(oplane + IMG_HW) + hwq);
        __builtin_nontemporal_store(ob4, (v4f*)(oplane + 2 * IMG_HW) + hwq);
    }
}

// ---------------------------------------------------------------------------
extern "C" void kernel_launch(void* const* d_in, const int* in_sizes, int n_in,
                              void* d_out, int out_size, void* d_ws, size_t ws_size,
                              hipStream_t stream) {
    const float* lut = (const float*)d_in[0];   // (3, 33, 33, 33) fp32
    const float* x   = (const float*)d_in[1];   // (8, 3, 1080, 1920) fp32
    float* out       = (float*)d_out;           // (8, 3, 1080, 1920) fp32

    const int use_ws = (d_ws != nullptr && ws_size >= RG_BYTES) ? 1 : 0;
    float2* rg = (float2*)d_ws;

    if (use_ws) {
        lut_prepack_kernel<<<(LUT_N + 255) / 256, 256, 0, stream>>>(lut, rg);
    }

    // Raise dynamic-LDS cap (capture-safe host API; ignore failure).
    static_cast<void>(hipFuncSetAttribute(
        reinterpret_cast<const void*>(&lut3d_apply_kernel),
        hipFuncAttributeMaxDynamicSharedMemorySize, (int)RG_BYTES));

    // One WG per WGP (287.5 KB LDS), 1024 threads = 32 wave32 waves per WGP.
    lut3d_apply_kernel<<<256, 1024, RG_BYTES, stream>>>(lut, x, out, rg, use_ws);
}